// ResMambaBlock_27779848470955
// MI455X (gfx1250) — compile-verified
//
#include <hip/hip_runtime.h>
#include <hip/hip_bf16.h>
#include <math.h>

typedef float v2f __attribute__((ext_vector_type(2)));
typedef float v8f __attribute__((ext_vector_type(8)));

#define CDIM 64
#define DIN  128
#define NST  16
#define KCV  4
#define BAT  4
#define LSEQ 9216
#define NCH  96
#define CHL  96
#define BL   (BAT*LSEQ)   /* 36864 */

__device__ __forceinline__ float sigmoidf_(float x){ return 1.f/(1.f+__expf(-x)); }

// ---------------- instance-norm statistics (per (b,c) over L) ----------------
__global__ __launch_bounds__(256) void k_in_stats(const float* __restrict__ x,
                                                  float* __restrict__ mean,
                                                  float* __restrict__ rstd){
  __shared__ float ss[256], sq[256];
  int bc = blockIdx.x;
  const float* p = x + (size_t)bc*LSEQ;
  float s=0.f, q=0.f;
  for (int i=threadIdx.x;i<LSEQ;i+=256){ float v=p[i]; s+=v; q+=v*v; }
  ss[threadIdx.x]=s; sq[threadIdx.x]=q; __syncthreads();
  for (int o=128;o>0;o>>=1){
    if (threadIdx.x<o){ ss[threadIdx.x]+=ss[threadIdx.x+o]; sq[threadIdx.x]+=sq[threadIdx.x+o]; }
    __syncthreads();
  }
  if (threadIdx.x==0){
    float m = ss[0]*(1.f/LSEQ);
    float v = sq[0]*(1.f/LSEQ) - m*m;
    mean[bc]=m; rstd[bc]=rsqrtf(fmaxf(v,0.f)+1e-5f);
  }
}

// --------- apply instance-norm + ReLU, transpose BCHW -> [B,L,C] -------------
__global__ __launch_bounds__(256) void k_in_apply(const float* __restrict__ x,
    const float* __restrict__ mean, const float* __restrict__ rstd,
    float* __restrict__ xf){
  int gid = blockIdx.x*256+threadIdx.x;
  if (gid >= BAT*CDIM*LSEQ) return;
  int l = gid % LSEQ; int bc = gid / LSEQ;
  int c = bc % CDIM;  int b  = bc / CDIM;
  float v = (x[gid]-mean[bc])*rstd[bc];
  v = fmaxf(v, 0.f);
  xf[((size_t)(b*LSEQ+l))*CDIM + c] = v;
}

// ---------------- LayerNorm over C=64 (one wave32 per row) -------------------
__global__ __launch_bounds__(256) void k_ln(const float* __restrict__ xf,
    const float* __restrict__ g, const float* __restrict__ be,
    float* __restrict__ xn){
  int wid  = (blockIdx.x*blockDim.x+threadIdx.x)>>5;
  int lane = threadIdx.x & 31;
  if (wid >= BL) return;
  const float* r = xf + (size_t)wid*CDIM;
  float v0 = r[lane], v1 = r[lane+32];
  float s = v0+v1, q = v0*v0+v1*v1;
  for (int o=16;o;o>>=1){ s+=__shfl_xor(s,o,32); q+=__shfl_xor(q,o,32); }
  float m   = s*(1.f/CDIM);
  float var = q*(1.f/CDIM)-m*m;
  float rs  = rsqrtf(fmaxf(var,0.f)+1e-5f);
  float* w = xn + (size_t)wid*CDIM;
  w[lane]    = (v0-m)*rs*g[lane]    + be[lane];
  w[lane+32] = (v1-m)*rs*g[lane+32] + be[lane+32];
}

// ------------- WMMA GEMM: out = A[M,K] @ W[N,K]^T, fused epilogues ----------
// mode 0: plain store           mode 1: softplus(acc + p0[n])  (dt path)
// mode 2: acc + p1[0]*p0[m,n]   (out_proj + skip*xf)
// mode 3: acc + p0[n], transposed store to [B,C,L]
// mode 4: mode 3 + p1 (identity residual) added
__global__ __launch_bounds__(256) void k_gemm(const float* __restrict__ A, int lda,
    const float* __restrict__ Wm, int ldw, int Kdim, int M, int Nact, int tnCnt,
    float* __restrict__ out, int ldout, int mode,
    const float* __restrict__ p0, const float* __restrict__ p1){
  int wave  = (blockIdx.x*blockDim.x+threadIdx.x)>>5;
  int tiles = (M>>4)*tnCnt;
  if (wave >= tiles) return;                 // wave-uniform: EXEC all-ones at WMMA
  int tm = wave / tnCnt, tn = wave % tnCnt;
  int lane = threadIdx.x & 31;
  int half = lane>>4, lm = lane&15, kb = half*2;
  const float* Ap = A + (size_t)(tm*16+lm)*lda + kb;
  int wcol = tn*16+lm;
  bool wok = wcol < Nact;
  const float* Wp = Wm + (size_t)(wok?wcol:0)*ldw + kb;
  v8f acc = {0.f,0.f,0.f,0.f,0.f,0.f,0.f,0.f};
  for (int k0=0;k0<Kdim;k0+=4){
    v2f av, wv;
    av.x = Ap[k0]; av.y = Ap[k0+1];
    float w0 = Wp[k0], w1 = Wp[k0+1];
    wv.x = wok ? w0 : 0.f; wv.y = wok ? w1 : 0.f;
    acc = __builtin_amdgcn_wmma_f32_16x16x4_f32(false, av, false, wv,
                                                (short)0, acc, false, false);
  }
  int n = tn*16 + lm;
  if (n >= Nact) return;
#pragma unroll
  for (int i=0;i<8;i++){
    int m = tm*16 + i + half*8;
    float v = acc[i];
    if (mode==0){
      out[(size_t)m*ldout + n] = v;
    } else if (mode==1){
      float t = v + p0[n];
      out[(size_t)m*ldout + n] = fmaxf(t,0.f) + log1pf(__expf(-fabsf(t)));
    } else if (mode==2){
      out[(size_t)m*ldout + n] = v + p1[0]*p0[(size_t)m*ldout + n];
    } else {
      int b = m / LSEQ, lr = m % LSEQ;
      size_t idx = ((size_t)b*CDIM + n)*LSEQ + lr;
      float r = v + p0[n];
      if (mode==4) r += p1[idx];
      out[idx] = r;
    }
  }
}

// ----------- depthwise causal conv (K=4) + SiLU on xin = xz[:, :DIN] ---------
__global__ __launch_bounds__(256) void k_conv(const float* __restrict__ xz,
    const float* __restrict__ cw, const float* __restrict__ cb,
    float* __restrict__ xc){
  int gid = blockIdx.x*256+threadIdx.x;
  if (gid >= BL*DIN) return;
  int d = gid % DIN; int bl = gid / DIN;
  int l = bl % LSEQ; int b  = bl / LSEQ;
  float acc = cb[d];
#pragma unroll
  for (int k=0;k<KCV;k++){
    int ls = l - (KCV-1) + k;
    if (ls >= 0) acc += xz[((size_t)(b*LSEQ+ls))*(2*DIN) + d] * cw[d*KCV+k];
  }
  xc[gid] = acc * sigmoidf_(acc);
}

// -------- scan pass 1: per-chunk affine composition (Aprod, Bcomb) -----------
__global__ __launch_bounds__(256) void k_scan_chunks(const float* __restrict__ dtf,
    const float* __restrict__ xc, const float* __restrict__ dbl,
    const float* __restrict__ A_log, float* __restrict__ chA, float* __restrict__ chB){
  int gid = blockIdx.x*256+threadIdx.x;
  if (gid >= BAT*NCH*DIN) return;
  int d = gid % DIN; int t = gid / DIN;
  int ch = t % NCH;  int b = t / NCH;
  float An[NST], Ap[NST], Bc[NST];
#pragma unroll
  for (int n=0;n<NST;n++){ An[n] = -__expf(A_log[d*NST+n]); Ap[n]=1.f; Bc[n]=0.f; }
  int l0 = ch*CHL;
  for (int i=0;i<CHL;i++){
    size_t bl = (size_t)b*LSEQ + l0 + i;
    float dt = dtf[bl*DIN+d];
    float xv = xc[bl*DIN+d];
    const float* bm = dbl + bl*36 + 4;
    float dx = dt*xv;
#pragma unroll
    for (int n=0;n<NST;n++){
      float a = __expf(dt*An[n]);
      Ap[n] *= a;
      Bc[n] = fmaf(a, Bc[n], dx*bm[n]);
    }
  }
  size_t o = ((size_t)(b*DIN+d)*NCH + ch)*NST;
#pragma unroll
  for (int n=0;n<NST;n++){ chA[o+n]=Ap[n]; chB[o+n]=Bc[n]; }
}

// -------- scan pass 2: serial prefix over the 96 chunks ----------------------
__global__ __launch_bounds__(256) void k_scan_prefix(const float* __restrict__ chA,
    const float* __restrict__ chB, float* __restrict__ hp){
  int gid = blockIdx.x*256+threadIdx.x;
  if (gid >= BAT*DIN*NST) return;
  int n = gid & (NST-1);
  size_t base = (size_t)(gid>>4)*NCH*NST + n;
  float h = 0.f;
  for (int ch=0;ch<NCH;ch++){
    size_t idx = base + (size_t)ch*NST;
    hp[idx] = h;
    h = fmaf(chA[idx], h, chB[idx]);
  }
}

// -------- scan pass 3: replay chunk, fuse h·C + D*xc and SiLU(z) gate --------
__global__ __launch_bounds__(256) void k_scan_apply(const float* __restrict__ dtf,
    const float* __restrict__ xc, const float* __restrict__ dbl,
    const float* __restrict__ A_log, const float* __restrict__ hp,
    const float* __restrict__ xz, const float* __restrict__ Dp,
    float* __restrict__ yb){
  int gid = blockIdx.x*256+threadIdx.x;
  if (gid >= BAT*NCH*DIN) return;
  int d = gid % DIN; int t = gid / DIN;
  int ch = t % NCH;  int b = t / NCH;
  float An[NST], h[NST];
  size_t o = ((size_t)(b*DIN+d)*NCH + ch)*NST;
#pragma unroll
  for (int n=0;n<NST;n++){ An[n] = -__expf(A_log[d*NST+n]); h[n]=hp[o+n]; }
  float Dv = Dp[d];
  int l0 = ch*CHL;
  for (int i=0;i<CHL;i++){
    size_t bl = (size_t)b*LSEQ + l0 + i;
    float dt = dtf[bl*DIN+d];
    float xv = xc[bl*DIN+d];
    const float* bm = dbl + bl*36 + 4;
    const float* cm = dbl + bl*36 + 20;
    float dx = dt*xv;
    float acc = 0.f;
#pragma unroll
    for (int n=0;n<NST;n++){
      float a = __expf(dt*An[n]);
      h[n] = fmaf(a, h[n], dx*bm[n]);
      acc  = fmaf(h[n], cm[n], acc);
    }
    float y = fmaf(Dv, xv, acc);
    float z = xz[bl*(2*DIN) + DIN + d];
    yb[bl*DIN+d] = y * (z * sigmoidf_(z));
  }
}

// ---------------------------------------------------------------------------
static void launch_layer(const float* xin, const float* const* P,
                         float* outb, const float* ident,
                         float* stats, float* xf, float* xn, float* xz,
                         float* xc, float* dblb, float* dtf,
                         float* chA, float* chB, float* hp,
                         float* yb, float* ymc, hipStream_t stream){
  float* mean = stats; float* rstd = stats + BAT*CDIM;
  k_in_stats<<<BAT*CDIM,256,0,stream>>>(xin, mean, rstd);
  k_in_apply<<<(BAT*CDIM*LSEQ+255)/256,256,0,stream>>>(xin, mean, rstd, xf);
  k_ln<<<BL/8,256,0,stream>>>(xf, P[0], P[1], xn);
  // in_proj: [BL,64] @ [256,64]^T -> xz [BL,256]
  k_gemm<<<((BL/16)*16+7)/8,256,0,stream>>>(xn, CDIM, P[2], CDIM, CDIM,
      BL, 2*DIN, 16, xz, 2*DIN, 0, nullptr, nullptr);
  k_conv<<<(BL*DIN+255)/256,256,0,stream>>>(xz, P[3], P[4], xc);
  // x_proj: [BL,128] @ [36,128]^T -> dbl [BL,36]
  k_gemm<<<((BL/16)*3+7)/8,256,0,stream>>>(xc, DIN, P[5], DIN, DIN,
      BL, 36, 3, dblb, 36, 0, nullptr, nullptr);
  // dt_proj + softplus: [BL,4] @ [128,4]^T -> dtf [BL,128]
  k_gemm<<<((BL/16)*8+7)/8,256,0,stream>>>(dblb, 36, P[6], 4, 4,
      BL, DIN, 8, dtf, DIN, 1, P[7], nullptr);
  k_scan_chunks<<<(BAT*NCH*DIN)/256,256,0,stream>>>(dtf, xc, dblb, P[8], chA, chB);
  k_scan_prefix<<<(BAT*DIN*NST)/256,256,0,stream>>>(chA, chB, hp);
  k_scan_apply<<<(BAT*NCH*DIN)/256,256,0,stream>>>(dtf, xc, dblb, P[8], hp, xz, P[9], yb);
  // out_proj + skip*xf: [BL,128] @ [64,128]^T -> ymc [BL,64]
  k_gemm<<<((BL/16)*4+7)/8,256,0,stream>>>(yb, DIN, P[10], DIN, DIN,
      BL, CDIM, 4, ymc, CDIM, 2, xf, P[11]);
  // pr proj + bias, transposed store to BCHW (+identity on layer 2)
  k_gemm<<<((BL/16)*4+7)/8,256,0,stream>>>(ymc, CDIM, P[12], CDIM, CDIM,
      BL, CDIM, 4, outb, 0, ident?4:3, P[13], ident);
}

extern "C" void kernel_launch(void* const* d_in, const int* in_sizes, int n_in,
                              void* d_out, int out_size, void* d_ws, size_t ws_size,
                              hipStream_t stream){
  (void)in_sizes; (void)n_in; (void)out_size; (void)ws_size;
  const float* x = (const float*)d_in[0];
  const float* P1[14]; const float* P2[14];
  for (int i=0;i<14;i++){ P1[i]=(const float*)d_in[1+i]; P2[i]=(const float*)d_in[15+i]; }
  float* w = (float*)d_ws;
  size_t off = 0;
  auto take = [&](size_t nf){ float* p = w + off; off += nf; return p; };
  float* stats = take(2*BAT*CDIM);
  float* xf  = take((size_t)BL*CDIM);
  float* xn  = take((size_t)BL*CDIM);
  float* xz  = take((size_t)BL*2*DIN);
  float* xc  = take((size_t)BL*DIN);
  float* dblb= take((size_t)BL*36);
  float* dtf = take((size_t)BL*DIN);
  float* chA = take((size_t)BAT*DIN*NCH*NST);
  float* chB = take((size_t)BAT*DIN*NCH*NST);
  float* hp  = take((size_t)BAT*DIN*NCH*NST);
  float* yb  = take((size_t)BL*DIN);
  float* ymc = take((size_t)BL*CDIM);
  float* xmid= take((size_t)BAT*CDIM*LSEQ);

  launch_layer(x,    P1, xmid,           nullptr, stats, xf, xn, xz, xc, dblb, dtf, chA, chB, hp, yb, ymc, stream);
  launch_layer(xmid, P2, (float*)d_out,  x,       stats, xf, xn, xz, xc, dblb, dtf, chA, chB, hp, yb, ymc, stream);
}